// RRoiPooling_35794257445125
// MI455X (gfx1250) — compile-verified
//
#include <hip/hip_runtime.h>

typedef __attribute__((ext_vector_type(2))) float v2f;
typedef __attribute__((ext_vector_type(4))) float v4f;
typedef __attribute__((ext_vector_type(8))) float v8f;

#define CH     128
#define FH     64
#define FW     96
#define NBINS  9
#define NTASK  54   /* 9 bins * 6 sample rows */

__global__ __launch_bounds__(128)
void psroi_gather_wmma(const float* __restrict__ fm,
                       const float* __restrict__ rois,
                       float* __restrict__ out)
{
    const int n    = blockIdx.x;
    const int tid  = threadIdx.x;
    const int wave = tid >> 5;
    const int lane = tid & 31;

    // ---- per-roi scalars (wave-uniform) ----
    const float* r = rois + n * 5;
    const int   b   = (int)r[0];
    const float x1n = r[1] / 1536.0f;   // IM_W
    const float y1n = r[2] / 1024.0f;   // IM_H
    const float x2n = r[3] / 1536.0f;
    const float y2n = r[4] / 1024.0f;
    const float bin_w = (x2n - x1n) / 3.0f;
    const float bin_h = y2n - y1n / 3.0f;          // reference precedence quirk

    const int clane = lane * 4;                    // this lane's channel base
    v4f acc = {0.f, 0.f, 0.f, 0.f};

    // ---- gather phase: 54 (bin,row) tasks strided over 4 waves ----
    for (int t = wave; t < NTASK; t += 4) {
        const int bin = t / 6;
        const int sy  = t - bin * 6;
        const int ih  = bin / 3;
        const int iw  = bin - ih * 3;

        const float by1 = y1n + (float)ih * bin_h;
        const float by2 = y1n + (float)(ih + 1) * bin_h;
        const float bx1 = x1n + (float)iw * bin_w;
        const float bx2 = x1n + (float)(iw + 1) * bin_h;   // reference quirk: bin_h

        const float ty = ((by2 - by1) * 63.0f) / 5.0f;
        const float ys = by1 * 63.0f + (float)sy * ty;
        if (!(ys >= 0.0f && ys <= 63.0f)) continue;        // masked row

        const float y0f = floorf(ys);
        const float ly  = ys - y0f;
        int y0 = (int)y0f; y0 = y0 < 0 ? 0 : (y0 > 63 ? 63 : y0);
        int y1 = y0 + 1;   y1 = y1 > 63 ? 63 : y1;

        const float tx = ((bx2 - bx1) * 95.0f) / 5.0f;
        const float xb = bx1 * 95.0f;

        const int rb = b * (FH * FW);
        const int r0 = (rb + y0 * FW) * (NBINS * CH) + bin * CH + clane;
        const int r1 = (rb + y1 * FW) * (NBINS * CH) + bin * CH + clane;

        #pragma unroll
        for (int sx = 0; sx < 6; ++sx) {
            const float xs = xb + (float)sx * tx;
            if (!(xs >= 0.0f && xs <= 95.0f)) continue;    // masked column

            const float x0f = floorf(xs);
            const float lx  = xs - x0f;
            int x0  = (int)x0f; x0 = x0 < 0 ? 0 : (x0 > 95 ? 95 : x0);
            int x1i = x0 + 1;   x1i = x1i > 95 ? 95 : x1i;

            const int o0 = x0  * (NBINS * CH);
            const int o1 = x1i * (NBINS * CH);

            const v4f v00 = *(const v4f*)(fm + r0 + o0);   // coalesced b128 gathers
            const v4f v01 = *(const v4f*)(fm + r0 + o1);
            const v4f v10 = *(const v4f*)(fm + r1 + o0);
            const v4f v11 = *(const v4f*)(fm + r1 + o1);

            const float omlx = 1.0f - lx;
            const float omly = 1.0f - ly;
            const v4f top = v00 * omlx + v01 * lx;         // match reference combine order
            const v4f bot = v10 * omlx + v11 * lx;
            acc += top * omly + bot * ly;
        }
    }

    // ---- stage the 4 per-wave partials in LDS ----
    __shared__ float lds[4 * CH];
    #pragma unroll
    for (int j = 0; j < 4; ++j) lds[wave * CH + clane + j] = acc[j];
    __syncthreads();

    // ---- reduce across waves with V_WMMA_F32_16X16X4_F32 ----
    // A = ones(16x4), B[k][n] = partial_k[cbase+n]  =>  D[m][n] = sum_k partial_k[cbase+n]
    // (invariant to the hardware's K-slot ordering inside B since A is all ones)
    const int   nn    = lane & 15;
    const int   hi    = lane >> 4;
    const float scale = 1.0f / 324.0f;   // /9 bins, /36 grid mean

    #pragma unroll
    for (int g = 0; g < 2; ++g) {
        const int cbase = wave * 32 + g * 16;   // waves cover channels 0..127
        v2f a; a.x = 1.0f; a.y = 1.0f;
        v2f bm;
        bm.x = lds[(2 * hi + 0) * CH + cbase + nn];
        bm.y = lds[(2 * hi + 1) * CH + cbase + nn];
        v8f c = {};
        c = __builtin_amdgcn_wmma_f32_16x16x4_f32(false, a, false, bm,
                                                  (short)0, c, false, false);
        if (lane < 16)
            out[n * CH + cbase + lane] = c[0] * scale;
    }
}

extern "C" void kernel_launch(void* const* d_in, const int* in_sizes, int n_in,
                              void* d_out, int out_size, void* d_ws, size_t ws_size,
                              hipStream_t stream)
{
    (void)n_in; (void)out_size; (void)d_ws; (void)ws_size;
    const float* fm   = (const float*)d_in[0];   // (4,64,96,1152) f32
    const float* rois = (const float*)d_in[1];   // (2048,5) f32
    float*       out  = (float*)d_out;           // (1,2048,128) f32
    const int n_rois = in_sizes[1] / 5;
    hipLaunchKernelGGL(psroi_gather_wmma, dim3(n_rois), dim3(128), 0, stream,
                       fm, rois, out);
}